// EfficientInteractionBilinear_86234353369150
// MI455X (gfx1250) — compile-verified
//
#include <hip/hip_runtime.h>

// ---------------------------------------------------------------------------
// EfficientInteractionBilinear on MI455X (gfx1250), fp32 via V_WMMA_F32_16X16X4_F32
//
// Shapes: E=30000 edges, K=8, S=16, I=64, H=128, O=128.
// Fused per-workgroup pipeline over 16 edges using ~304KB of the 320KB WGP LDS:
//   sum_k^T  : [16e][128h][16s pad 18]  = 147,456 B
//   X chunk  : [16e][16i' pad][128h]     (stride 2084 f)  = 133,376 B
//   rbf chunk: [16e][16i'][16s pad 18]   =  18,432 B
//   m2 stage : [8k][128h pad 132]        =   4,224 B
//   sph      : [128]                     =     512 B
// ---------------------------------------------------------------------------

typedef __attribute__((ext_vector_type(2))) float v2f;
typedef __attribute__((ext_vector_type(8))) float v8f;

#define SPH    16
#define INTERM 64
#define EMB    128
#define OUT_U  128
#define EB     16          // edges per workgroup (== WMMA M for the bilinear GEMM)

// LDS partition (float offsets); strides padded for bank-conflict-free frags
#define OFF_SUMKT 0
#define SUMKT_E   (128 * 18)        // 2304 floats per edge, [h][s] (transposed)
#define OFF_X     36864
#define XSTR_I    130               // i'-row stride inside an edge
#define XSTR_E    2084              // 16*130 + 4  (lane-e banks distinct: 2084%64=36)
#define OFF_RBF   70208
#define RBF_E     288               // 16 * 18
#define OFF_M2    74816             // 8 * 132
#define OFF_SPH   75872
#define SMEM_FLOATS 76000           // 304,000 bytes <= 320KB WGP LDS

__global__ void k_map_init(int* __restrict__ map, int n) {
    int i = blockIdx.x * 256 + threadIdx.x;
    if (i < n) map[i] = -1;
}

__global__ void k_map_scatter(const int* __restrict__ id_reduce,
                              const int* __restrict__ id_ragged,
                              int* __restrict__ map, int K, int nt) {
    int t = blockIdx.x * 256 + threadIdx.x;
    if (t < nt) map[(size_t)id_reduce[t] * K + id_ragged[t]] = t;
}

__global__ __launch_bounds__(256)
void k_bilinear(const float* __restrict__ rbf,     // (E, 64, 16)
                const float* __restrict__ sph,     // (E, 16, K)
                const float* __restrict__ m,       // (NT, 128)
                const float* __restrict__ weight,  // (128, 64, 128)
                const int*   __restrict__ map,     // (E, K) -> triplet row or -1
                float* __restrict__ out,           // (E, 128)
                int E, int K)
{
    extern __shared__ float sm[];
    const int tid  = threadIdx.x;
    const int lane = tid & 31;         // wave32
    const int w    = tid >> 5;         // 8 waves; wave w owns 16-wide h/o tile w
    const int half = lane >> 4;        // K-half of the fragment (rows 0,1 vs 2,3)
    const int ln   = lane & 15;        // M row (A) / N col (B,C,D)
    const int jo   = half * 2;         // K offset inside a 16x16x4 step
    const int e0   = blockIdx.x * EB;

    // ------------------- Stage 1: sum_k^T per edge -------------------------
    for (int eL = 0; eL < EB; ++eL) {
        const int e = e0 + eL;
        if (tid < SPH * K)
            sm[OFF_SPH + tid] = (e < E) ? sph[(size_t)e * SPH * K + tid] : 0.f;
        // ragged gather of K rows of m (float4 granularity)
        const int tot = K * EMB;
        for (int base = tid * 4; base < tot; base += 256 * 4) {
            const int k  = base >> 7;
            const int h0 = base & 127;
            float4 v = make_float4(0.f, 0.f, 0.f, 0.f);
            if (e < E) {
                const int t = map[(size_t)e * K + k];
                if (t >= 0) v = *(const float4*)(m + (size_t)t * EMB + h0);
            }
            float* d = sm + OFF_M2 + k * 132 + h0;
            d[0] = v.x; d[1] = v.y; d[2] = v.z; d[3] = v.w;
        }
        __syncthreads();
        // sum_k[s][h] = sph[s][k] * m2[k][h], stored transposed [h][s pad 18]
        {
            const int h  = tid >> 1;
            const int s0 = (tid & 1) * 8;
            float acc[8];
#pragma unroll
            for (int j = 0; j < 8; ++j) acc[j] = 0.f;
            for (int k = 0; k < K; ++k) {
                const float mv = sm[OFF_M2 + k * 132 + h];
#pragma unroll
                for (int j = 0; j < 8; ++j)
                    acc[j] += sm[OFF_SPH + (s0 + j) * K + k] * mv;
            }
            float* d = sm + OFF_SUMKT + eL * SUMKT_E + h * 18 + s0;
#pragma unroll
            for (int j = 0; j < 8; ++j) d[j] = acc[j];
        }
        __syncthreads();
    }

    // ------------------- Stage 2: i-chunked WMMA pipeline -------------------
    v8f accC = {};                                  // out[16 edges][o-tile w]
    for (int ic = 0; ic < INTERM / 16; ++ic) {
        const int i0 = ic * 16;
        // load rbf chunk: [16e][16i'][16s pad 18]
#pragma unroll
        for (int r = 0; r < 16; ++r) {
            const int lin = r * 256 + tid;
            const int eL = lin >> 8, rem = lin & 255;
            const int ip = rem >> 4, s  = rem & 15;
            const int ge = e0 + eL;
            const float v = (ge < E)
                ? rbf[((size_t)ge * INTERM + i0 + ip) * SPH + s] : 0.f;
            sm[OFF_RBF + eL * RBF_E + ip * 18 + s] = v;
        }
        __syncthreads();

        // 2b: X[eL](16i' x 128h) = rbf_chunk(16x16) @ sum_k(16x128); wave w -> h-tile w
        for (int eL = 0; eL < EB; ++eL) {
            const float* ra = sm + OFF_RBF  + eL * RBF_E  + ln * 18 + jo;
            const float* rb = sm + OFF_SUMKT + eL * SUMKT_E + (w * 16 + ln) * 18 + jo;
            v8f acc = {};
#pragma unroll
            for (int kb = 0; kb < 16; kb += 4) {
                v2f a = *(const v2f*)(ra + kb);     // A: rows=i', K=s
                v2f b = *(const v2f*)(rb + kb);     // B: rows=s (transposed store)
                acc = __builtin_amdgcn_wmma_f32_16x16x4_f32(
                          false, a, false, b, (short)0, acc, false, false);
            }
            float* xd = sm + OFF_X + eL * XSTR_E + w * 16 + ln;
#pragma unroll
            for (int r = 0; r < 8; ++r)
                xd[(half * 8 + r) * XSTR_I] = acc[r];
        }
        __syncthreads();

        // 2c: accC(16e x 16o) += X(16e x 2048) @ W(2048 x 16o); W L2-resident
        {
            const float* xa = sm + OFF_X + ln * XSTR_E;   // A row = edge = ln
            const int o0 = w * 16;
            for (int c = 0; c < 16 * EMB; c += 4) {
                const int ip = c >> 7;                    // i' within chunk
                const int h0 = c & 127;                   // h base (4 rows same i')
                v2f a = *(const v2f*)(xa + ip * XSTR_I + h0 + jo);
                const float* wp = weight
                    + (((size_t)(h0 + jo) * INTERM + (i0 + ip)) << 7) + o0 + ln;
                v2f b;
                b.x = wp[0];
                b.y = wp[(size_t)INTERM * OUT_U];         // next h row: +8192 floats
                accC = __builtin_amdgcn_wmma_f32_16x16x4_f32(
                           false, a, false, b, (short)0, accC, false, false);
            }
        }
        __syncthreads();
    }

    // ------------------- Epilogue: D rows are edges -------------------------
    {
        const int o0 = w * 16;
#pragma unroll
        for (int r = 0; r < 8; ++r) {
            const int er = e0 + half * 8 + r;
            if (er < E) out[(size_t)er * OUT_U + o0 + ln] = accC[r];
        }
    }
}

extern "C" void kernel_launch(void* const* d_in, const int* in_sizes, int n_in,
                              void* d_out, int out_size, void* d_ws, size_t ws_size,
                              hipStream_t stream)
{
    (void)n_in; (void)out_size; (void)ws_size;
    const float* rbf = (const float*)d_in[0];   // (E, 64, 16)
    const float* sph = (const float*)d_in[1];   // (E, 16, K)
    const float* m   = (const float*)d_in[2];   // (NT, 128)
    const float* wt  = (const float*)d_in[3];   // (128, 64, 128)
    const int*   idr = (const int*)d_in[4];     // (NT)
    const int*   idk = (const int*)d_in[5];     // (NT)
    // d_in[6] is Kmax on device; derive shapes on host instead:
    const int E  = in_sizes[0] / (INTERM * SPH);
    const int K  = in_sizes[1] / (E * SPH);     // == 8
    const int NT = in_sizes[2] / EMB;

    int* map = (int*)d_ws;                      // E*K ints (~0.96 MB)
    const int nmap = E * K;
    hipLaunchKernelGGL(k_map_init, dim3((nmap + 255) / 256), dim3(256), 0, stream,
                       map, nmap);
    hipLaunchKernelGGL(k_map_scatter, dim3((NT + 255) / 256), dim3(256), 0, stream,
                       idr, idk, map, K, NT);

    const int nblk = (E + EB - 1) / EB;         // 1875
    hipLaunchKernelGGL(k_bilinear, dim3(nblk), dim3(256),
                       SMEM_FLOATS * sizeof(float), stream,
                       rbf, sph, m, wt, map, (float*)d_out, E, K);
}